// SGAT_25159918420558
// MI455X (gfx1250) — compile-verified
//
#include <hip/hip_runtime.h>
#include <hip/hip_bf16.h>
#include <stdint.h>

typedef float v2f __attribute__((ext_vector_type(2)));
typedef float v8f __attribute__((ext_vector_type(8)));

#define N_NODES 50000
#define N_EDGES 800000
#define IN_DIM 128
#define HIDDEN 128
#define NUM_CLASSES 64
#define NEG_SLOPE 0.2f

// ============================================================================
// GEMM: C[M x (NT*16)] = A[M x 128] @ B[128 x (NT*16)], fp32 via WMMA 16x16x4.
// One wave computes a 16-row stripe across all NT column tiles.
// ============================================================================
template <int NT>
__global__ __launch_bounds__(256) void gemm_f32_wmma(
    const float* __restrict__ A, const float* __restrict__ B,
    float* __restrict__ C, int M) {
  constexpr int K = 128;
  constexpr int N = NT * 16;
  __shared__ float Bs[K * N];  // row-major, conflict-free reads
  for (int i = threadIdx.x; i < K * N; i += 256) Bs[i] = B[i];
  __syncthreads();

  const int wave = threadIdx.x >> 5;
  const int lane = threadIdx.x & 31;
  const int mt = blockIdx.x * 8 + wave;      // wave-uniform
  if (mt * 16 >= M) return;                  // whole wave exits -> EXEC all 1s
  const int m0 = mt * 16;
  const int ml = lane & 15;                  // row within tile
  const int kb = (lane >> 4) << 1;           // lanes 16-31 carry K+2,K+3

  const float* Arow = A + (size_t)(m0 + ml) * K;
  v8f acc[NT] = {};
  for (int k = 0; k < K; k += 4) {
    v2f a;                                   // A 16x4 f32 fragment (ISA 7.12.2)
    a.x = Arow[k + kb];
    a.y = Arow[k + kb + 1];
#pragma unroll
    for (int j = 0; j < NT; ++j) {
      v2f b;                                 // B 4x16 fragment
      b.x = Bs[(k + kb) * N + j * 16 + ml];
      b.y = Bs[(k + kb + 1) * N + j * 16 + ml];
      acc[j] = __builtin_amdgcn_wmma_f32_16x16x4_f32(
          /*neg_a=*/false, a, /*neg_b=*/false, b,
          /*c_mod=*/(short)0, acc[j], /*reuse_a=*/false, /*reuse_b=*/false);
    }
  }
  // D layout: VGPR r -> row r (lanes 0-15) / row r+8 (lanes 16-31), col = ml
  const int mr = m0 + ((lane >> 4) << 3);
#pragma unroll
  for (int j = 0; j < NT; ++j)
#pragma unroll
    for (int r = 0; r < 8; ++r)
      C[(size_t)(mr + r) * N + j * 16 + ml] = acc[j][r];
}

// ============================================================================
// el/er = h @ attn_l , h @ attn_r   (one wave per node, float4 + shfl reduce)
// ============================================================================
__global__ __launch_bounds__(256) void gat_scores(
    const float* __restrict__ h, const float* __restrict__ al,
    const float* __restrict__ ar, float* __restrict__ el,
    float* __restrict__ er, int n) {
  const int wave = threadIdx.x >> 5, lane = threadIdx.x & 31;
  const int node = blockIdx.x * 8 + wave;
  if (node >= n) return;
  float4 v = ((const float4*)(h + (size_t)node * HIDDEN))[lane];
  float4 a = ((const float4*)al)[lane];
  float4 b = ((const float4*)ar)[lane];
  float sl = v.x * a.x + v.y * a.y + v.z * a.z + v.w * a.w;
  float sr = v.x * b.x + v.y * b.y + v.z * b.z + v.w * b.w;
  for (int off = 16; off; off >>= 1) {
    sl += __shfl_down(sl, off, 32);
    sr += __shfl_down(sr, off, 32);
  }
  if (lane == 0) { el[node] = sl; er[node] = sr; }
}

// ---------- monotone uint key for float atomic-max ----------
__device__ __forceinline__ unsigned fkey(float f) {
  unsigned b = __float_as_uint(f);
  return (b & 0x80000000u) ? ~b : (b | 0x80000000u);
}
__device__ __forceinline__ float funkey(unsigned k) {
  return __uint_as_float((k & 0x80000000u) ? (k & 0x7FFFFFFFu) : ~k);
}

__global__ void node_init(unsigned* __restrict__ mk, float* __restrict__ esum, int n) {
  int t = blockIdx.x * blockDim.x + threadIdx.x;
  if (t < n) { mk[t] = fkey(-__builtin_inff()); esum[t] = 0.f; }
}

// pass A: e = leaky_relu(el[src]+er[dst]); segment max via atomicMax(key)
__global__ void edge_score(const int* __restrict__ src, const int* __restrict__ dst,
                           const float* __restrict__ el, const float* __restrict__ er,
                           float* __restrict__ e, unsigned* __restrict__ mk, int E) {
  int t = blockIdx.x * blockDim.x + threadIdx.x;
  if (t >= E) return;
  int s = src[t], d = dst[t];
  float v = el[s] + er[d];
  v = v > 0.f ? v : NEG_SLOPE * v;
  e[t] = v;
  atomicMax(&mk[d], fkey(v));
}

// pass B: ee = exp(e - max[dst]); segment sum
__global__ void edge_exp(const int* __restrict__ dst, const unsigned* __restrict__ mk,
                         float* __restrict__ e, float* __restrict__ esum, int E) {
  int t = blockIdx.x * blockDim.x + threadIdx.x;
  if (t >= E) return;
  int d = dst[t];
  float ee = __expf(e[t] - funkey(mk[d]));
  e[t] = ee;
  atomicAdd(&esum[d], ee);
}

// pass C: alpha = ee / sum[dst]   (in place)
__global__ void edge_norm(const int* __restrict__ dst, const float* __restrict__ esum,
                          float* __restrict__ e, int E) {
  int t = blockIdx.x * blockDim.x + threadIdx.x;
  if (t >= E) return;
  e[t] = e[t] / esum[dst[t]];
}

// out[n*F + c] = bias[c] (or 0 if bias==nullptr)
__global__ void fill_bias(float* __restrict__ out, const float* __restrict__ bias,
                          int total, int F) {
  int t = blockIdx.x * blockDim.x + threadIdx.x;
  if (t < total) out[t] = bias ? bias[t % F] : 0.f;
}

__global__ void elu_inplace(float* __restrict__ x, int total) {
  int t = blockIdx.x * blockDim.x + threadIdx.x;
  if (t < total) {
    float v = x[t];
    x[t] = v > 0.f ? v : (__expf(v) - 1.f);
  }
}

// ============================================================================
// SpMM: out[dst] += hp[src] * alpha.   F/4 lanes per edge, float4 per lane,
// wave-uniform src/dst/alpha for F=128 (one wave per edge). L2-resident gather.
// ============================================================================
template <int F>
__global__ __launch_bounds__(256) void spmm_atomic(
    const float* __restrict__ hp, const float* __restrict__ alpha,
    const int* __restrict__ src, const int* __restrict__ dst,
    float* __restrict__ out, int E) {
  constexpr int PER = F / 4;
  long long t = (long long)blockIdx.x * blockDim.x + threadIdx.x;
  int e = (int)(t / PER);
  if (e >= E) return;
  int c = (int)(t % PER) * 4;
  int s = src[e], d = dst[e];
  float a = alpha[e];
  float4 v = *(const float4*)(hp + (size_t)s * F + c);
  float* o = out + (size_t)d * F + c;
  atomicAdd(o + 0, v.x * a);
  atomicAdd(o + 1, v.y * a);
  atomicAdd(o + 2, v.z * a);
  atomicAdd(o + 3, v.w * a);
}

// ============================================================================
extern "C" void kernel_launch(void* const* d_in, const int* in_sizes, int n_in,
                              void* d_out, int out_size, void* d_ws, size_t ws_size,
                              hipStream_t stream) {
  (void)in_sizes; (void)n_in; (void)out_size; (void)ws_size;
  const float* x      = (const float*)d_in[0];
  const int*   src    = (const int*)d_in[1];
  const int*   dst    = (const int*)d_in[2];
  const float* W_gat  = (const float*)d_in[3];
  const float* attn_l = (const float*)d_in[4];
  const float* attn_r = (const float*)d_in[5];
  const float* W1     = (const float*)d_in[6];
  const float* b1     = (const float*)d_in[7];
  const float* W2     = (const float*)d_in[8];
  const float* b2     = (const float*)d_in[9];
  const float* Wc     = (const float*)d_in[10];
  const float* bc     = (const float*)d_in[11];
  float* logits = (float*)d_out;

  // ---- workspace layout ----
  char* w = (char*)d_ws;
  auto take = [&](size_t bytes) {
    char* p = w;
    w += (bytes + 255) & ~(size_t)255;
    return p;
  };
  float*    hA   = (float*)take((size_t)N_NODES * HIDDEN * 4);
  float*    hB   = (float*)take((size_t)N_NODES * HIDDEN * 4);
  float*    el   = (float*)take((size_t)N_NODES * 4);
  float*    er   = (float*)take((size_t)N_NODES * 4);
  unsigned* mk   = (unsigned*)take((size_t)N_NODES * 4);
  float*    esum = (float*)take((size_t)N_NODES * 4);
  float*    ebuf = (float*)take((size_t)N_EDGES * 4);

  const int MT = (N_NODES + 15) / 16;          // 3125 row tiles
  const int GEMM_BLKS = (MT + 7) / 8;          // 8 waves/block
  const int EB = (N_EDGES + 255) / 256;
  const int NB = (N_NODES + 255) / 256;
  const int NH = N_NODES * HIDDEN;
  const int NHB = (NH + 255) / 256;
  const int NC = N_NODES * NUM_CLASSES;

  // 1) h = x @ W_gat
  gemm_f32_wmma<8><<<GEMM_BLKS, 256, 0, stream>>>(x, W_gat, hA, N_NODES);
  // 2) el/er
  gat_scores<<<(N_NODES + 7) / 8, 256, 0, stream>>>(hA, attn_l, attn_r, el, er, N_NODES);
  // 3) edge softmax
  node_init<<<NB, 256, 0, stream>>>(mk, esum, N_NODES);
  edge_score<<<EB, 256, 0, stream>>>(src, dst, el, er, ebuf, mk, N_EDGES);
  edge_exp<<<EB, 256, 0, stream>>>(dst, mk, ebuf, esum, N_EDGES);
  edge_norm<<<EB, 256, 0, stream>>>(dst, esum, ebuf, N_EDGES);
  // 4) h1 = elu(segment_sum(h[src]*alpha))
  fill_bias<<<NHB, 256, 0, stream>>>(hB, nullptr, NH, HIDDEN);
  spmm_atomic<HIDDEN><<<(int)(((long long)N_EDGES * (HIDDEN / 4) + 255) / 256), 256, 0,
                        stream>>>(hA, ebuf, src, dst, hB, N_EDGES);
  elu_inplace<<<NHB, 256, 0, stream>>>(hB, NH);
  // 5) gconv1: h2 = seg_sum((h1@W1)[src]*alpha) + b1
  gemm_f32_wmma<8><<<GEMM_BLKS, 256, 0, stream>>>(hB, W1, hA, N_NODES);
  fill_bias<<<NHB, 256, 0, stream>>>(hB, b1, NH, HIDDEN);
  spmm_atomic<HIDDEN><<<(int)(((long long)N_EDGES * (HIDDEN / 4) + 255) / 256), 256, 0,
                        stream>>>(hA, ebuf, src, dst, hB, N_EDGES);
  // 6) gconv2
  gemm_f32_wmma<8><<<GEMM_BLKS, 256, 0, stream>>>(hB, W2, hA, N_NODES);
  fill_bias<<<NHB, 256, 0, stream>>>(hB, b2, NH, HIDDEN);
  spmm_atomic<HIDDEN><<<(int)(((long long)N_EDGES * (HIDDEN / 4) + 255) / 256), 256, 0,
                        stream>>>(hA, ebuf, src, dst, hB, N_EDGES);
  // 7) classifier gconv: logits = seg_sum((h3@Wc)[src]*alpha) + bc
  gemm_f32_wmma<4><<<GEMM_BLKS, 256, 0, stream>>>(hB, Wc, hA, N_NODES);
  fill_bias<<<(NC + 255) / 256, 256, 0, stream>>>(logits, bc, NC, NUM_CLASSES);
  spmm_atomic<NUM_CLASSES><<<(int)(((long long)N_EDGES * (NUM_CLASSES / 4) + 255) / 256),
                             256, 0, stream>>>(hA, ebuf, src, dst, logits, N_EDGES);
}